// Net_18348100288785
// MI455X (gfx1250) — compile-verified
//
#include <hip/hip_runtime.h>
#include <math.h>

#define NN 100000
#define EE 1600000
#define CC 128
#define HH 64
#define GG 512

#define S1 132   // LDS row stride for 16x128 tile (pad to avoid bank conflicts; mult of 4 -> 16B aligned rows)
#define S2 68    // LDS row stride for 16x64 tile

typedef __attribute__((ext_vector_type(2))) float v2f;
typedef __attribute__((ext_vector_type(8))) float v8f;

// ---------------- helpers ----------------

__device__ __forceinline__ void atomicMaxF(float* addr, float v) {
  // int ordering == float ordering for non-negative; reversed unsigned for negative
  if (v >= 0.0f) atomicMax((int*)addr, __float_as_int(v));
  else           atomicMin((unsigned int*)addr, __float_as_uint(v));
}

// ---------------- init ----------------

__global__ void k_zero(float* __restrict__ p, long long n4) {
  long long i = (long long)blockIdx.x * blockDim.x + threadIdx.x;
  if (i < n4) ((float4*)p)[i] = make_float4(0.f, 0.f, 0.f, 0.f);
}

__global__ void k_init_max(float* __restrict__ gmax, float* __restrict__ smax) {
  int i = blockIdx.x * blockDim.x + threadIdx.x;
  if (i < GG) { gmax[i] = -3.402823466e38f; smax[i] = -3.402823466e38f; }
}

// ---------------- pooling scores ----------------

// raw = x @ pool_w ; per-graph max (one wave32 per node)
__global__ __launch_bounds__(256) void k_raw(const float* __restrict__ x,
                                             const float* __restrict__ pw,
                                             const int* __restrict__ batch,
                                             float* __restrict__ score,
                                             float* __restrict__ gmax) {
  int wid  = (int)(((long long)blockIdx.x * blockDim.x + threadIdx.x) >> 5);
  int lane = threadIdx.x & 31;
  if (wid >= NN) return;
  const float* xr = x + (size_t)wid * CC;
  float s = 0.0f;
#pragma unroll
  for (int c = lane; c < CC; c += 32) s += xr[c] * pw[c];
#pragma unroll
  for (int off = 16; off > 0; off >>= 1) s += __shfl_xor(s, off, 32);
  if (lane == 0) { score[wid] = s; atomicMaxF(&gmax[batch[wid]], s); }
}

__global__ void k_exp(float* __restrict__ score, const float* __restrict__ gmax,
                      const int* __restrict__ batch, float* __restrict__ z) {
  int i = blockIdx.x * blockDim.x + threadIdx.x;
  if (i >= NN) return;
  int g = batch[i];
  float e = __expf(score[i] - gmax[g]);
  score[i] = e;
  atomicAdd(&z[g], e);
}

__global__ void k_norm(float* __restrict__ score, const float* __restrict__ z,
                       const int* __restrict__ batch, float* __restrict__ smax) {
  int i = blockIdx.x * blockDim.x + threadIdx.x;
  if (i >= NN) return;
  int g = batch[i];
  float s = score[i] / z[g];
  score[i] = s;
  atomicMaxF(&smax[g], s);
}

__global__ void k_mask(const float* __restrict__ score, const float* __restrict__ smax,
                       const int* __restrict__ batch, const float* __restrict__ attn,
                       float* __restrict__ maskf, float* __restrict__ sm,
                       float* __restrict__ klsum, float* __restrict__ cnt) {
  int i = blockIdx.x * blockDim.x + threadIdx.x;
  if (i >= NN) return;
  int g = batch[i];
  float s = score[i];
  float thr = fminf(smax[g] - 1e-7f, 0.05f);
  float m = (s > thr) ? 1.0f : 0.0f;
  maskf[i] = m;
  sm[i] = s * m;
  if (m != 0.0f) {
    float a  = attn[i];
    float kl = a * (__logf(a) - __logf(s + 1e-14f));
    atomicAdd(&klsum[g], kl);
    atomicAdd(&cnt[g], 1.0f);
  }
}

// ---------------- edge aggregations (float4 granularity) ----------------

// thread per (edge, 4-channel chunk): b128 gather, 4 scalar atomic adds
__global__ __launch_bounds__(256) void k_agg1(const float* __restrict__ x,
                                              const int* __restrict__ ei,
                                              float* __restrict__ agg) {
  long long idx = (long long)blockIdx.x * blockDim.x + threadIdx.x;
  if (idx >= (long long)EE * (CC / 4)) return;
  int c4 = (int)(idx & (CC / 4 - 1));
  int e  = (int)(idx >> 5);
  int s = ei[e], d = ei[EE + e];
  float4 v = ((const float4*)x)[(size_t)s * (CC / 4) + c4];
  float* dst = &agg[(size_t)d * CC + c4 * 4];
  atomicAdd(dst + 0, v.x);
  atomicAdd(dst + 1, v.y);
  atomicAdd(dst + 2, v.z);
  atomicAdd(dst + 3, v.w);
}

__global__ __launch_bounds__(256) void k_agg2(const float* __restrict__ outp,
                                              const int* __restrict__ ei,
                                              const float* __restrict__ maskf,
                                              float* __restrict__ agg) {
  long long idx = (long long)blockIdx.x * blockDim.x + threadIdx.x;
  if (idx >= (long long)EE * (HH / 4)) return;
  int c4 = (int)(idx & (HH / 4 - 1));
  int e  = (int)(idx >> 4);
  int s = ei[e], d = ei[EE + e];
  float w = maskf[s] * maskf[d];
  if (w != 0.0f) {
    float4 v = ((const float4*)outp)[(size_t)s * (HH / 4) + c4];
    float* dst = &agg[(size_t)d * HH + c4 * 4];
    atomicAdd(dst + 0, v.x * w);
    atomicAdd(dst + 1, v.y * w);
    atomicAdd(dst + 2, v.z * w);
    atomicAdd(dst + 3, v.w * w);
  }
}

// ---------------- WMMA GIN MLPs ----------------
// 4 waves per block; wave w computes the 16x16 output tile for columns [16w,16w+16).
// A fragment (16x4 f32): lane l -> row m=l&15, K pair kb = (l>>4)*2 (ISA 7.12.2).
// B fragment (4x16 f32): lane l -> col n=l&15, K rows kb, kb+1 (mirror of A).
// C/D (16x16 f32): VGPR v -> row v + 8*(l>>4), col l&15.

__global__ __launch_bounds__(128) void k_mlp1(const float* __restrict__ x,
                                              const float* __restrict__ agg,
                                              const float* __restrict__ W1,
                                              const float* __restrict__ b1,
                                              const float* __restrict__ W2,
                                              const float* __restrict__ b2,
                                              const float* __restrict__ sm,
                                              float* __restrict__ outp) {
  __shared__ float hA[16 * S1];
  __shared__ float hB[16 * S2];
  const int base = blockIdx.x * 16;
  const int tid  = threadIdx.x;
  const int wave = tid >> 5, lane = tid & 31;

  // stage h = x + agg as float4 (b128 loads, b128 LDS stores)
  for (int i = tid; i < 16 * (CC / 4); i += 128) {
    int r = i >> 5, c4 = i & (CC / 4 - 1);
    size_t gi = (size_t)(base + r) * (CC / 4) + c4;
    float4 xv = ((const float4*)x)[gi];
    float4 av = ((const float4*)agg)[gi];
    float4 h  = make_float4(xv.x + av.x, xv.y + av.y, xv.z + av.z, xv.w + av.w);
    *(float4*)&hA[r * S1 + c4 * 4] = h;
  }
  __syncthreads();

  const int m    = lane & 15;
  const int kb   = (lane >> 4) << 1;
  const int ncol = (wave << 4) + m;

  v8f acc = {};
  for (int k0 = 0; k0 < CC; k0 += 4) {
    v2f a, b;
    a.x = hA[m * S1 + k0 + kb];
    a.y = hA[m * S1 + k0 + kb + 1];
    b.x = W1[(k0 + kb) * HH + ncol];
    b.y = W1[(k0 + kb + 1) * HH + ncol];
    acc = __builtin_amdgcn_wmma_f32_16x16x4_f32(false, a, false, b, (short)0, acc, false, false);
  }
  const int rb = (lane >> 4) << 3;
  float bias = b1[ncol];
#pragma unroll
  for (int v = 0; v < 8; ++v)
    hB[(rb + v) * S2 + ncol] = fmaxf(acc[v] + bias, 0.0f);
  __syncthreads();

  v8f acc2 = {};
  for (int k0 = 0; k0 < HH; k0 += 4) {
    v2f a, b;
    a.x = hB[m * S2 + k0 + kb];
    a.y = hB[m * S2 + k0 + kb + 1];
    b.x = W2[(k0 + kb) * HH + ncol];
    b.y = W2[(k0 + kb + 1) * HH + ncol];
    acc2 = __builtin_amdgcn_wmma_f32_16x16x4_f32(false, a, false, b, (short)0, acc2, false, false);
  }
  float bias2 = b2[ncol];
#pragma unroll
  for (int v = 0; v < 8; ++v) {
    int node = base + rb + v;
    outp[(size_t)node * HH + ncol] = fmaxf(acc2[v] + bias2, 0.0f) * sm[node];
  }
}

__global__ __launch_bounds__(128) void k_mlp2(const float* __restrict__ outp,
                                              const float* __restrict__ agg,
                                              const float* __restrict__ W3,
                                              const float* __restrict__ b3,
                                              const float* __restrict__ W4,
                                              const float* __restrict__ b4,
                                              const float* __restrict__ maskf,
                                              const int* __restrict__ batch,
                                              float* __restrict__ gout) {
  __shared__ float hA[16 * S2];
  __shared__ float hB[16 * S2];
  const int base = blockIdx.x * 16;
  const int tid  = threadIdx.x;
  const int wave = tid >> 5, lane = tid & 31;

  for (int i = tid; i < 16 * (HH / 4); i += 128) {
    int r = i >> 4, c4 = i & (HH / 4 - 1);
    size_t gi = (size_t)(base + r) * (HH / 4) + c4;
    float4 ov = ((const float4*)outp)[gi];
    float4 av = ((const float4*)agg)[gi];
    float4 h  = make_float4(ov.x + av.x, ov.y + av.y, ov.z + av.z, ov.w + av.w);
    *(float4*)&hA[r * S2 + c4 * 4] = h;
  }
  __syncthreads();

  const int m    = lane & 15;
  const int kb   = (lane >> 4) << 1;
  const int ncol = (wave << 4) + m;

  v8f acc = {};
  for (int k0 = 0; k0 < HH; k0 += 4) {
    v2f a, b;
    a.x = hA[m * S2 + k0 + kb];
    a.y = hA[m * S2 + k0 + kb + 1];
    b.x = W3[(k0 + kb) * HH + ncol];
    b.y = W3[(k0 + kb + 1) * HH + ncol];
    acc = __builtin_amdgcn_wmma_f32_16x16x4_f32(false, a, false, b, (short)0, acc, false, false);
  }
  const int rb = (lane >> 4) << 3;
  float bias = b3[ncol];
#pragma unroll
  for (int v = 0; v < 8; ++v)
    hB[(rb + v) * S2 + ncol] = fmaxf(acc[v] + bias, 0.0f);
  __syncthreads();

  v8f acc2 = {};
  for (int k0 = 0; k0 < HH; k0 += 4) {
    v2f a, b;
    a.x = hB[m * S2 + k0 + kb];
    a.y = hB[m * S2 + k0 + kb + 1];
    b.x = W4[(k0 + kb) * HH + ncol];
    b.y = W4[(k0 + kb + 1) * HH + ncol];
    acc2 = __builtin_amdgcn_wmma_f32_16x16x4_f32(false, a, false, b, (short)0, acc2, false, false);
  }
  float bias2 = b4[ncol];
#pragma unroll
  for (int v = 0; v < 8; ++v) {
    int node = base + rb + v;
    float val = fmaxf(acc2[v] + bias2, 0.0f) * maskf[node];
    atomicAdd(&gout[(size_t)batch[node] * HH + ncol], val);
  }
}

// ---------------- readout ----------------

__global__ __launch_bounds__(256) void k_final(const float* __restrict__ gout,
                                               const float* __restrict__ Wl,
                                               const float* __restrict__ bl,
                                               const float* __restrict__ klsum,
                                               const float* __restrict__ cnt,
                                               float* __restrict__ out) {
  int wid  = (int)(((long long)blockIdx.x * blockDim.x + threadIdx.x) >> 5);
  int lane = threadIdx.x & 31;
  if (wid < GG) {
    const float* gr = gout + (size_t)wid * HH;
    float s = gr[lane] * Wl[lane] + gr[lane + 32] * Wl[lane + 32];
#pragma unroll
    for (int off = 16; off > 0; off >>= 1) s += __shfl_xor(s, off, 32);
    if (lane == 0) {
      out[wid]      = s + bl[0];
      out[GG + wid] = klsum[wid] / fmaxf(cnt[wid], 1.0f);
    }
  }
  if (blockIdx.x == 0 && threadIdx.x == 0) {
    float t = 0.0f;
    for (int g = 0; g < GG; ++g) t += cnt[g];
    out[2 * GG] = t / (float)NN;
  }
}

// ---------------- launcher ----------------

extern "C" void kernel_launch(void* const* d_in, const int* in_sizes, int n_in,
                              void* d_out, int out_size, void* d_ws, size_t ws_size,
                              hipStream_t stream) {
  (void)in_sizes; (void)n_in; (void)out_size; (void)ws_size;

  const float* x     = (const float*)d_in[0];
  const int*   ei    = (const int*)  d_in[1];
  const int*   batch = (const int*)  d_in[2];
  const float* attn  = (const float*)d_in[3];
  const float* W1    = (const float*)d_in[4];
  const float* b1    = (const float*)d_in[5];
  const float* W2    = (const float*)d_in[6];
  const float* b2    = (const float*)d_in[7];
  const float* pw    = (const float*)d_in[8];
  const float* W3    = (const float*)d_in[9];
  const float* b3    = (const float*)d_in[10];
  const float* W4    = (const float*)d_in[11];
  const float* b4    = (const float*)d_in[12];
  const float* Wl    = (const float*)d_in[13];
  const float* bl    = (const float*)d_in[14];
  float* out = (float*)d_out;

  float* ws = (float*)d_ws;
  size_t off = 0;
  float* agg1  = ws + off; off += (size_t)NN * CC;   // zeroed
  float* agg2  = ws + off; off += (size_t)NN * HH;   // zeroed
  float* gout  = ws + off; off += (size_t)GG * HH;   // zeroed
  float* z     = ws + off; off += GG;                // zeroed
  float* klsum = ws + off; off += GG;                // zeroed
  float* cnt   = ws + off; off += GG;                // zeroed
  size_t zeroFloats = off;
  float* score = ws + off; off += NN;
  float* sm    = ws + off; off += NN;
  float* maskf = ws + off; off += NN;
  float* gmax  = ws + off; off += GG;
  float* smax  = ws + off; off += GG;
  float* outp  = ws + off; off += (size_t)NN * HH;

  long long n4 = (long long)(zeroFloats / 4);
  k_zero<<<(unsigned)((n4 + 255) / 256), 256, 0, stream>>>(ws, n4);
  k_init_max<<<2, 256, 0, stream>>>(gmax, smax);

  k_raw<<<(NN * 32 + 255) / 256, 256, 0, stream>>>(x, pw, batch, score, gmax);
  k_exp<<<(NN + 255) / 256, 256, 0, stream>>>(score, gmax, batch, z);
  k_norm<<<(NN + 255) / 256, 256, 0, stream>>>(score, z, batch, smax);
  k_mask<<<(NN + 255) / 256, 256, 0, stream>>>(score, smax, batch, attn, maskf, sm, klsum, cnt);

  k_agg1<<<(unsigned)(((long long)EE * (CC / 4) + 255) / 256), 256, 0, stream>>>(x, ei, agg1);
  k_mlp1<<<NN / 16, 128, 0, stream>>>(x, agg1, W1, b1, W2, b2, sm, outp);

  k_agg2<<<(unsigned)(((long long)EE * (HH / 4) + 255) / 256), 256, 0, stream>>>(outp, ei, maskf, agg2);
  k_mlp2<<<NN / 16, 128, 0, stream>>>(outp, agg2, W3, b3, W4, b4, maskf, batch, gout);

  k_final<<<(GG * 32) / 256, 256, 0, stream>>>(gout, Wl, bl, klsum, cnt, out);
}